// Segmenter_SAPS_19877108646375
// MI455X (gfx1250) — compile-verified
//
#include <hip/hip_runtime.h>
#include <hip/hip_bf16.h>

typedef __attribute__((ext_vector_type(2))) float v2f;
typedef __attribute__((ext_vector_type(8))) float v8f;

// Padded LDS row stride (dwords) for T so B-fragment half-wave loads
// (rows k0 / k0+2) land on disjoint bank sets: 528 % 64 = 16 banks shift/row.
#define T_STRIDE 528

__global__ __launch_bounds__(256) void saps_bilinear_wmma_kernel(
    const float* __restrict__ scores,          // [4, 1024, 150] f32
    const long long* __restrict__ patch_index, // [32, 32] int64 (identity here)
    float* __restrict__ out)                   // [4, 150, 512, 512] f32
{
    constexpr int N = 1024, C = 150, G = 32, O = 512;

    __shared__ float Ms[G * G];        // gathered 32x32 class-score tile
    __shared__ float Ts[G * T_STRIDE]; // x-interpolated 32x512 (padded)

    const int tid = threadIdx.x;
    const int blk = blockIdx.x;
    const int s   = blk & 1;           // output-row half (tiles 0-15 / 16-31)
    const int bc  = blk >> 1;          // 0..599
    const int b   = bc / C;
    const int c   = bc % C;

    // ---- Phase 0: gather M[h,w] = scores[b, patch_index[h,w], c] ----
    for (int n = tid; n < N; n += 256) {
        int p = (int)patch_index[n];
        Ms[n] = scores[(size_t)(b * N + p) * C + c];
    }
    __syncthreads();

    // ---- Phase 1: x-direction bilinear (half-pixel coords, edge clamp) ----
    // T[h,x] = (1-f)*M[h,w0] + f*M[h,w1],  src = (x+0.5)/16 - 0.5 = (2x-15)/32
    for (int i = tid; i < G * O; i += 256) {
        int h = i >> 9;
        int x = i & (O - 1);
        float src = (float)(2 * x - 15) * (1.0f / 32.0f);
        int w0 = (int)floorf(src);
        w0 = min(max(w0, 0), G - 1);
        float f = fminf(fmaxf(src - (float)w0, 0.0f), 1.0f);
        int w1 = min(w0 + 1, G - 1);
        float v0 = Ms[h * G + w0];
        float v1 = Ms[h * G + w1];
        Ts[h * T_STRIDE + x] = v0 + f * (v1 - v0);
    }
    __syncthreads();

    // ---- Phase 2: y-direction as WMMA f32 16x16x4 ----
    // Out tile (ty,tx) = Ry[16ty:+16, k0:+4] x T[k0:+4, 16tx:+16], one wmma each.
    const int wave = tid >> 5;
    const int lane = tid & 31;
    const int hi   = lane >> 4;   // half-wave select per ISA VGPR layout
    const int ln   = lane & 15;

    float* outbc = out + ((size_t)bc << 18); // 512*512 = 1<<18

    for (int tyi = 0; tyi < 2; ++tyi) {
        int ty = s * 16 + wave + 8 * tyi;        // tile row 0..31
        int k0 = min(max(ty - 1, 0), G - 4);     // 4 source rows cover this tile

        // A fragment: A[m,k] = Ry[16ty+m, k0+k]; lane ln = m.
        // VGPR0 holds K = hi?2:0, VGPR1 holds K = hi?3:1 (ISA 16x4 f32 layout).
        int y = 16 * ty + ln;
        float src = (float)(2 * y - 15) * (1.0f / 32.0f);
        int h0 = (int)floorf(src);
        h0 = min(max(h0, 0), G - 1);
        float f = fminf(fmaxf(src - (float)h0, 0.0f), 1.0f);
        int h1 = min(h0 + 1, G - 1);
        int ka = k0 + (hi ? 2 : 0);
        int kb = k0 + (hi ? 3 : 1);
        v2f a;
        a.x = (ka == h0 ? 1.0f - f : 0.0f) + (ka == h1 ? f : 0.0f);
        a.y = (kb == h0 ? 1.0f - f : 0.0f) + (kb == h1 ? f : 0.0f);

        for (int tx = 0; tx < 32; ++tx) {
            int xg = 16 * tx + ln;
            // B fragment: B[k,n] = T[k0+k, 16tx+n]; row striped across lanes.
            v2f bf;
            bf.x = Ts[(k0 + (hi ? 2 : 0)) * T_STRIDE + xg];
            bf.y = Ts[(k0 + (hi ? 3 : 1)) * T_STRIDE + xg];
            v8f cz = {};
            v8f d = __builtin_amdgcn_wmma_f32_16x16x4_f32(
                false, a, false, bf, (short)0, cz, false, false);
            // D layout: VGPR r -> row r (lanes 0-15) / r+8 (lanes 16-31), col = ln.
            size_t rowbase = (size_t)(16 * ty + hi * 8) * O + (size_t)xg;
            #pragma unroll
            for (int r = 0; r < 8; ++r) {
                __builtin_nontemporal_store(d[r], &outbc[rowbase + (size_t)r * O]);
            }
        }
    }
}

extern "C" void kernel_launch(void* const* d_in, const int* in_sizes, int n_in,
                              void* d_out, int out_size, void* d_ws, size_t ws_size,
                              hipStream_t stream) {
    (void)in_sizes; (void)n_in; (void)d_ws; (void)ws_size; (void)out_size;
    const float*     scores = (const float*)d_in[0];
    const long long* pidx   = (const long long*)d_in[1];
    float*           out    = (float*)d_out;

    // B*C*2 = 4*150*2 = 1200 workgroups, 256 threads (8 wave32) each.
    dim3 grid(1200), block(256);
    saps_bilinear_wmma_kernel<<<grid, block, 0, stream>>>(scores, pidx, out);
}